// SIAB_85899346585
// MI455X (gfx1250) — compile-verified
//
#include <hip/hip_runtime.h>
#include <hip/hip_bf16.h>

// Problem constants (from reference)
#define B_    64
#define C_    128
#define HW_   4096          // 64*64
#define NDOM  4
#define NPD   16            // samples per domain
#define EPSV  1e-5f

typedef __attribute__((ext_vector_type(2))) float v2f;
typedef __attribute__((ext_vector_type(8))) float v8f;

// Accumulate sum of two f32 values per lane (one 16x4 A-matrix per wave = 64 elems)
// into a 16x16 f32 accumulator via V_WMMA_F32_16X16X4_F32 with B = all-ones.
// With B==1, sum over ALL entries of D equals 16 * (sum of all A entries),
// independent of VGPR striping details.
__device__ __forceinline__ void wmma_sum(v8f& acc, float a0, float a1, v2f ones) {
    v2f a; a.x = a0; a.y = a1;
    acc = __builtin_amdgcn_wmma_f32_16x16x4_f32(
        /*neg_a=*/false, a, /*neg_b=*/false, ones,
        /*c_mod=*/(short)0, acc, /*reuse_a=*/false, /*reuse_b=*/false);
}

// Full-wave (32 lane) total of the WMMA accumulator; *0.0625 undoes the 16x
// replication from the all-ones B (exact: power-of-two scale).
__device__ __forceinline__ float wave_total(v8f acc) {
    float s = acc[0] + acc[1] + acc[2] + acc[3] + acc[4] + acc[5] + acc[6] + acc[7];
    #pragma unroll
    for (int off = 16; off >= 1; off >>= 1)
        s += __shfl_xor(s, off, 32);
    return s * 0.0625f;
}

// ---------------------------------------------------------------------------
// Kernel 1: per-(domain, channel) BN stats -> folded scale/shift.
// grid = 4*128 blocks, 256 threads. Each block reduces 16*4096 = 65536 floats.
// ---------------------------------------------------------------------------
__global__ void __launch_bounds__(256)
bn_stats_kernel(const float* __restrict__ x, const int* __restrict__ sort_idx,
                const float* __restrict__ bn_w, const float* __restrict__ bn_b,
                float* __restrict__ bn_scale, float* __restrict__ bn_shift) {
    const int d = blockIdx.x >> 7;     // 0..3
    const int c = blockIdx.x & 127;    // 0..127
    const int t = threadIdx.x;

    v2f ones; ones.x = 1.0f; ones.y = 1.0f;
    v8f accS = {};  // sum
    v8f accQ = {};  // sum of squares

    for (int i = 0; i < NPD; ++i) {
        const int j = sort_idx[d * NPD + i];
        const float4* __restrict__ p =
            (const float4*)(x + ((size_t)j * C_ + c) * HW_);
        #pragma unroll
        for (int k = 0; k < 4; ++k) {
            float4 v = p[t + k * 256];
            wmma_sum(accS, v.x, v.y, ones);
            wmma_sum(accQ, v.x * v.x, v.y * v.y, ones);
            wmma_sum(accS, v.z, v.w, ones);
            wmma_sum(accQ, v.z * v.z, v.w * v.w, ones);
        }
    }

    float s = wave_total(accS);
    float q = wave_total(accQ);

    __shared__ float ls[8], lq[8];
    const int wave = t >> 5, lane = t & 31;
    if (lane == 0) { ls[wave] = s; lq[wave] = q; }
    __syncthreads();

    if (t == 0) {
        float S = 0.f, Q = 0.f;
        #pragma unroll
        for (int w = 0; w < 8; ++w) { S += ls[w]; Q += lq[w]; }
        const float invN = 1.0f / 65536.0f;
        float mean = S * invN;
        float var  = Q * invN - mean * mean;
        float inv  = rsqrtf(var + EPSV);
        float sc   = inv * bn_w[d * C_ + c];
        bn_scale[d * C_ + c] = sc;
        bn_shift[d * C_ + c] = bn_b[d * C_ + c] - mean * sc;
    }
}

// ---------------------------------------------------------------------------
// Kernel 2: per-(sample-in-domain-3, channel) InstanceNorm stats.
// grid = 16*128 blocks, 256 threads. Each block reduces 4096 floats.
// Stats stored indexed by ORIGINAL sample index j = sort_idx[48+m].
// ---------------------------------------------------------------------------
__global__ void __launch_bounds__(256)
in_stats_kernel(const float* __restrict__ x, const int* __restrict__ sort_idx,
                const float* __restrict__ in_w, const float* __restrict__ in_b,
                float* __restrict__ in_scale, float* __restrict__ in_shift) {
    const int m = blockIdx.x >> 7;     // 0..15 (sample within last domain)
    const int c = blockIdx.x & 127;
    const int t = threadIdx.x;
    const int j = sort_idx[(NDOM - 1) * NPD + m];   // original sample index

    v2f ones; ones.x = 1.0f; ones.y = 1.0f;
    v8f accS = {};
    v8f accQ = {};

    const float4* __restrict__ p = (const float4*)(x + ((size_t)j * C_ + c) * HW_);
    #pragma unroll
    for (int k = 0; k < 4; ++k) {
        float4 v = p[t + k * 256];
        wmma_sum(accS, v.x, v.y, ones);
        wmma_sum(accQ, v.x * v.x, v.y * v.y, ones);
        wmma_sum(accS, v.z, v.w, ones);
        wmma_sum(accQ, v.z * v.z, v.w * v.w, ones);
    }

    float s = wave_total(accS);
    float q = wave_total(accQ);

    __shared__ float ls[8], lq[8];
    const int wave = t >> 5, lane = t & 31;
    if (lane == 0) { ls[wave] = s; lq[wave] = q; }
    __syncthreads();

    if (t == 0) {
        float S = 0.f, Q = 0.f;
        #pragma unroll
        for (int w = 0; w < 8; ++w) { S += ls[w]; Q += lq[w]; }
        const float invN = 1.0f / 4096.0f;
        float mean = S * invN;
        float var  = Q * invN - mean * mean;
        float inv  = rsqrtf(var + EPSV);
        float sc   = inv * in_w[c];
        in_scale[j * C_ + c] = sc;
        in_shift[j * C_ + c] = in_b[c] - mean * sc;
    }
}

// ---------------------------------------------------------------------------
// Kernel 3: streaming apply. out[j,c,:,:] = x*scale + shift, where for the
// last domain scale/shift are the sigmoid(alpha)-blend of BN and IN params.
// grid = 64*128 blocks (one (j,c) plane = 4096 floats), 256 threads.
// ---------------------------------------------------------------------------
__global__ void __launch_bounds__(256)
apply_kernel(const float* __restrict__ x, const int* __restrict__ unsort_idx,
             const float* __restrict__ bn_scale, const float* __restrict__ bn_shift,
             const float* __restrict__ in_scale, const float* __restrict__ in_shift,
             const float* __restrict__ alpha, float* __restrict__ out) {
    const int j = blockIdx.x >> 7;
    const int c = blockIdx.x & 127;
    const int t = threadIdx.x;

    const int d = unsort_idx[j] >> 4;   // sorted position / 16 = domain
    float sc = bn_scale[d * C_ + c];
    float sh = bn_shift[d * C_ + c];
    if (d == NDOM - 1) {
        float tt = 1.0f / (1.0f + __expf(-alpha[c]));
        sc = tt * sc + (1.0f - tt) * in_scale[j * C_ + c];
        sh = tt * sh + (1.0f - tt) * in_shift[j * C_ + c];
    }

    const float4* __restrict__ px = (const float4*)(x   + ((size_t)j * C_ + c) * HW_);
    float4*       __restrict__ po = (float4*)      (out + ((size_t)j * C_ + c) * HW_);
    #pragma unroll
    for (int k = 0; k < 4; ++k) {
        float4 v = px[t + k * 256];
        v.x = fmaf(v.x, sc, sh);
        v.y = fmaf(v.y, sc, sh);
        v.z = fmaf(v.z, sc, sh);
        v.w = fmaf(v.w, sc, sh);
        po[t + k * 256] = v;
    }
}

// ---------------------------------------------------------------------------
extern "C" void kernel_launch(void* const* d_in, const int* in_sizes, int n_in,
                              void* d_out, int out_size, void* d_ws, size_t ws_size,
                              hipStream_t stream) {
    const float* x        = (const float*)d_in[0];
    const int*   sort_idx = (const int*)  d_in[1];
    const int*   unsort   = (const int*)  d_in[2];
    const float* bn_w     = (const float*)d_in[3];
    const float* bn_b     = (const float*)d_in[4];
    const float* in_w     = (const float*)d_in[5];
    const float* in_b     = (const float*)d_in[6];
    const float* alpha    = (const float*)d_in[7];
    float* out = (float*)d_out;

    // Workspace layout (floats):
    //   [0      ..  512) bn_scale[4][128]
    //   [512    .. 1024) bn_shift[4][128]
    //   [1024   .. 9216) in_scale[64][128]
    //   [9216   ..17408) in_shift[64][128]
    float* ws       = (float*)d_ws;
    float* bn_scale = ws;
    float* bn_shift = ws + 512;
    float* in_scale = ws + 1024;
    float* in_shift = ws + 9216;

    bn_stats_kernel<<<NDOM * C_, 256, 0, stream>>>(x, sort_idx, bn_w, bn_b,
                                                   bn_scale, bn_shift);
    in_stats_kernel<<<NPD * C_, 256, 0, stream>>>(x, sort_idx, in_w, in_b,
                                                  in_scale, in_shift);
    apply_kernel<<<B_ * C_, 256, 0, stream>>>(x, unsort, bn_scale, bn_shift,
                                              in_scale, in_shift, alpha, out);
}